// PnP_contour_feature_67860483276931
// MI455X (gfx1250) — compile-verified
//
#include <hip/hip_runtime.h>
#include <hip/hip_bf16.h>

// ---------------------------------------------------------------------------
// PnP contour rasterization + feature fusion for MI455X (gfx1250, wave32).
//
// Kernel 1 (pnp_raster): per block = (one polygon) x (16 rows).
//   - async-copy 128 vertices (1 KB) to LDS (ASYNCcnt path)
//   - per edge: slope/intercept precompute (xint is affine in py)
//   - XINT[16 edges x 16 rows] tiles via v_wmma_f32_16x16x4_f32:
//       A[e,k] = {c0_e, slope_e, 0, 0},  B[k,r] = {1, py_r, 0, 0}
//   - epilogue applies the scanline-crossing predicate -> xeff in LDS
//   - raster loop: parity ^= (px < xeff[e]), 4 edges per ds_load_b128
// Kernel 2: gather mask -> mask_batch + channel max  (faithful starts[] rule)
// Kernel 3: cnn_out = relu(max_mask*feat + feat), float4 vectorized
// ---------------------------------------------------------------------------

#define P_NUM        40
#define N_EDGE       128
#define H_DIM        256
#define W_DIM        256
#define BS_DIM       8
#define CIN_DIM      64
#define ROWS_PER_BLK 16
#define XEFF_PITCH   132   // 132*4 = 528 bytes/row: 16B aligned, conflict-free

typedef __attribute__((ext_vector_type(2))) float v2f;
typedef __attribute__((ext_vector_type(8))) float v8f;
typedef __attribute__((ext_vector_type(4))) int   v4i;
typedef __attribute__((address_space(1))) v4i*    gv4i_p;  // global int4*
typedef __attribute__((address_space(3))) v4i*    lv4i_p;  // LDS int4*

__global__ __launch_bounds__(256) void pnp_raster_kernel(
    const float* __restrict__ contour,   // [P, N, 2]
    float* __restrict__ mask)            // [P, H, W]
{
  __shared__ float2 s_vert[N_EDGE];                    // (x, y)
  __shared__ float4 s_eparm[N_EDGE];                   // (c0, slope, y1, y2)
  __shared__ float  s_xeff[ROWS_PER_BLK * XEFF_PITCH]; // [row][edge] padded

  const int tid     = threadIdx.x;
  const int p       = blockIdx.y;
  const int rowBase = blockIdx.x * ROWS_PER_BLK;

  // ---- stage the polygon's vertices into LDS ------------------------------
#if defined(__AMDGCN__) && __has_builtin(__builtin_amdgcn_global_load_async_to_lds_b128)
  if (tid < 64) {  // 64 lanes x 16B = 1024B = 128 float2 vertices
    gv4i_p g = (gv4i_p)((__attribute__((address_space(1))) char*)(
        contour + (size_t)p * N_EDGE * 2) + tid * 16);
    lv4i_p l = (lv4i_p)((__attribute__((address_space(3))) char*)s_vert
        + tid * 16);
    __builtin_amdgcn_global_load_async_to_lds_b128(g, l, 0, 0);
  }
  #if __has_builtin(__builtin_amdgcn_s_wait_asynccnt)
  __builtin_amdgcn_s_wait_asynccnt(0);
  #else
  asm volatile("s_wait_asynccnt 0" ::: "memory");
  #endif
#else
  if (tid < N_EDGE) {
    s_vert[tid] = ((const float2*)contour)[(size_t)p * N_EDGE + tid];
  }
#endif
  __syncthreads();

  // ---- per-edge affine parameters:  xint(py) = c0 + slope*py --------------
  if (tid < N_EDGE) {
    float2 v0 = s_vert[tid];
    float2 v1 = s_vert[(tid + 1) & (N_EDGE - 1)];
    float denom = (v1.y == v0.y) ? 1.0f : (v1.y - v0.y);
    float slope = (v1.x - v0.x) / denom;
    float c0    = v0.x - slope * v0.y;
    s_eparm[tid] = make_float4(c0, slope, v0.y, v1.y);
  }
  __syncthreads();

  // ---- XINT[128 x 16] = A[128x2] @ B[2x16] via WMMA -----------------------
#if defined(__AMDGCN__) && __has_builtin(__builtin_amdgcn_wmma_f32_16x16x4_f32)
  {
    const int wave = tid >> 5;     // 8 waves -> 8 edge tiles of 16
    const int lane = tid & 31;
    v2f a = {};                    // A 16x4 f32: lanes 0-15 = K0,K1 (M=lane)
    v2f b = {};                    // B 4x16 f32: lanes 0-15 = K0,K1 (N=lane)
    if (lane < 16) {
      float4 ep = s_eparm[(wave << 4) + lane];
      a.x = ep.x;                    // c0_e        (K=0)
      a.y = ep.y;                    // slope_e     (K=1)
      b.x = 1.0f;                    // B[0][row]   (K=0)
      b.y = (float)(rowBase + lane); // py_row      (K=1)
    }                              // lanes 16-31 carry K=2,3 -> zero pad
    v8f c = {};
    v8f d = __builtin_amdgcn_wmma_f32_16x16x4_f32(
        false, a, false, b, (short)0, c, false, false);
    // D element: M = v + 8*(lane>=16) (edge in tile), N = lane&15 (row)
    const int   row   = lane & 15;
    const int   ehalf = (lane >> 4) << 3;
    const float py    = (float)(rowBase + row);
    #pragma unroll
    for (int v = 0; v < 8; ++v) {
      int edge = (wave << 4) + ehalf + v;
      float4 ep = s_eparm[edge];
      bool crosses = (ep.z > py) != (ep.w > py);
      s_xeff[row * XEFF_PITCH + edge] = crosses ? d[v] : -3.0e30f;
    }
  }
#else
  // VALU fallback: 2048 xeff entries, 8 per thread
  for (int k = tid; k < N_EDGE * ROWS_PER_BLK; k += 256) {
    int edge = k & (N_EDGE - 1);
    int row  = k >> 7;
    float py = (float)(rowBase + row);
    float4 ep = s_eparm[edge];
    bool crosses = (ep.z > py) != (ep.w > py);
    s_xeff[row * XEFF_PITCH + edge] = crosses ? fmaf(ep.y, py, ep.x) : -3.0e30f;
  }
#endif
  __syncthreads();

  // ---- raster: each thread owns one pixel column, walks 16 rows -----------
  const float fpx = (float)tid;
  size_t obase = ((size_t)p * H_DIM + rowBase) * W_DIM + tid;
  for (int r = 0; r < ROWS_PER_BLK; ++r) {
    // all 32 lanes of a wave read the same LDS address -> broadcast
    const float4* xe4 = (const float4*)&s_xeff[r * XEFF_PITCH];
    bool parity = false;
    #pragma unroll 8
    for (int e4 = 0; e4 < N_EDGE / 4; ++e4) {
      float4 q = xe4[e4];
      parity ^= (fpx < q.x);
      parity ^= (fpx < q.y);
      parity ^= (fpx < q.z);
      parity ^= (fpx < q.w);
    }
    mask[obase + (size_t)r * W_DIM] = parity ? 1.0f : 0.0f;
  }
}

// ---- gather mask channels per batch + running max ---------------------------
__global__ __launch_bounds__(256) void gather_max_kernel(
    const float* __restrict__ mask,      // [P, H, W]
    const int*   __restrict__ ct_num,    // [BS]
    float* __restrict__ mask_batch,      // [BS, C, H, W]
    float* __restrict__ max_mask,        // [BS, H, W]
    int C)
{
  const int b   = blockIdx.y;
  const int row = blockIdx.x;
  const int px  = threadIdx.x;
  const int cn    = ct_num[b];
  const int start = (b == 0) ? 0 : ct_num[b - 1];  // faithful: NOT a cumsum
  const size_t pix = (size_t)row * W_DIM + px;

  float mx = 0.0f;
  for (int j = 0; j < C; ++j) {
    int idx = start + j;
    idx = idx < 0 ? 0 : (idx > P_NUM - 1 ? P_NUM - 1 : idx);
    float v = (j < cn) ? mask[(size_t)idx * (H_DIM * W_DIM) + pix] : 0.0f;
    mask_batch[((size_t)b * C + j) * (H_DIM * W_DIM) + pix] = v;
    mx = fmaxf(mx, v);
  }
  max_mask[(size_t)b * (H_DIM * W_DIM) + pix] = mx;
}

// ---- cnn_out = relu(max_mask * feat + feat), float4-vectorized --------------
__global__ __launch_bounds__(256) void fuse_relu_kernel(
    const float* __restrict__ feat,      // [BS, CIN, H, W]
    const float* __restrict__ max_mask,  // [BS, H, W]
    float* __restrict__ out)             // [BS, CIN, H, W]
{
  const size_t i = (size_t)blockIdx.x * blockDim.x + threadIdx.x; // float4 id
  const size_t x4   = i & (W_DIM / 4 - 1);       // W/4 = 64
  const size_t t    = i >> 6;
  const size_t y    = t & (H_DIM - 1);
  const size_t bc   = t >> 8;                    // b*CIN + c
  const size_t b    = bc >> 6;                   // CIN = 64

  float4 f = ((const float4*)feat)[i];
  float4 m = ((const float4*)max_mask)[(b * H_DIM + y) * (W_DIM / 4) + x4];
  float4 o;
  o.x = fmaxf(fmaf(m.x, f.x, f.x), 0.0f);
  o.y = fmaxf(fmaf(m.y, f.y, f.y), 0.0f);
  o.z = fmaxf(fmaf(m.z, f.z, f.z), 0.0f);
  o.w = fmaxf(fmaf(m.w, f.w, f.w), 0.0f);
  ((float4*)out)[i] = o;
}

extern "C" void kernel_launch(void* const* d_in, const int* in_sizes, int n_in,
                              void* d_out, int out_size, void* d_ws, size_t ws_size,
                              hipStream_t stream) {
  const float* contour = (const float*)d_in[0];  // [40, 128, 2] f32
  const float* feat    = (const float*)d_in[1];  // [8, 64, 256, 256] f32
  const int*   ct_num  = (const int*)d_in[2];    // [8] i32
  float* out = (float*)d_out;

  // out = mask_batch [BS,C,H,W] ++ cnn_out [BS,CIN,H,W]; recover C from size
  int C = out_size / (BS_DIM * H_DIM * W_DIM) - CIN_DIM;
  if (C < 1)  C = 1;
  if (C > 16) C = 16;

  // workspace: mask (10.5 MB) + max_mask (2 MB)
  float* mask     = (float*)d_ws;
  float* max_mask = mask + (size_t)P_NUM * H_DIM * W_DIM;

  float* mask_batch = out;
  float* cnn_out    = out + (size_t)BS_DIM * C * H_DIM * W_DIM;

  pnp_raster_kernel<<<dim3(H_DIM / ROWS_PER_BLK, P_NUM), 256, 0, stream>>>(
      contour, mask);

  gather_max_kernel<<<dim3(H_DIM, BS_DIM), W_DIM, 0, stream>>>(
      mask, ct_num, mask_batch, max_mask, C);

  const int n4 = BS_DIM * CIN_DIM * H_DIM * W_DIM / 4;  // 8,388,608
  fuse_relu_kernel<<<n4 / 256, 256, 0, stream>>>(feat, max_mask, cnn_out);
}